// SwinTransformerBlock_18648747999834
// MI455X (gfx1250) — compile-verified
//
#include <hip/hip_runtime.h>
#include <hip/hip_bf16.h>
#include <math.h>

// ---- problem constants (fixed by the reference) ----
#define BB   2
#define CC   192
#define SS   16
#define HH   32
#define WW   32
#define NHD  6
#define PP   4
#define HD   32
#define HID  768
#define LL   16384            // SS*HH*WW
#define NPROJ 288             // 192 (value) + 72 (offsets) + 24 (attn logits)

typedef float v2f __attribute__((ext_vector_type(2)));
typedef float v8f __attribute__((ext_vector_type(8)));

// D = A(16x4) * B(4x16) + C, fp32 WMMA (wave32).
__device__ __forceinline__ v8f wmma4(v2f a, v2f b, v8f c) {
    return __builtin_amdgcn_wmma_f32_16x16x4_f32(
        /*neg_a=*/false, a, /*neg_b=*/false, b,
        /*c_mod=*/(short)0, c, /*reuse_a=*/false, /*reuse_b=*/false);
}

__device__ __forceinline__ v8f vzero8() {
    v8f z = {0.f,0.f,0.f,0.f,0.f,0.f,0.f,0.f};
    return z;
}

// ------------------------------------------------------------------
// K1: pack (Wv | Woff | Waw) -> Wcat[192][288], biases -> bcat[288]
// ------------------------------------------------------------------
__global__ __launch_bounds__(256) void k_pack(
        const float* __restrict__ Wv,  const float* __restrict__ bv,
        const float* __restrict__ Woff,const float* __restrict__ boff,
        const float* __restrict__ Waw, const float* __restrict__ baw,
        float* __restrict__ Wcat, float* __restrict__ bcat) {
    int i = blockIdx.x * 256 + threadIdx.x;
    if (i < CC * NPROJ) {
        int k = i / NPROJ, n = i % NPROJ;
        float v;
        if (n < 192)      v = Wv[k * 192 + n];
        else if (n < 264) v = Woff[k * 72 + (n - 192)];
        else              v = Waw[k * 24 + (n - 264)];
        Wcat[i] = v;
    }
    if (i < NPROJ) {
        bcat[i] = (i < 192) ? bv[i] : (i < 264 ? boff[i - 192] : baw[i - 264]);
    }
}

// ------------------------------------------------------------------
// K2: Y[b, l, 0:288] = xp[b,l,:] @ Wcat + bcat
// xp[b,l,c] = X[b*C*L + c*L + l] (channel-major source -> coalesced A loads)
// one wave = one 16-token tile x TWO 16-col tiles (A reused for both)
// ------------------------------------------------------------------
__global__ __launch_bounds__(256) void k_gemm_proj(
        const float* __restrict__ X, const float* __restrict__ Wcat,
        const float* __restrict__ bcat, float* __restrict__ Y) {
    const int lane = threadIdx.x & 31;
    const int wv   = threadIdx.x >> 5;
    const int half = lane >> 4;
    const int ml   = lane & 15;
    const int tileM = blockIdx.x * 8 + wv;          // 0 .. B*L/16-1
    const int n0    = blockIdx.y * 16;              // blockIdx.y in [0,9)
    const int n1    = n0 + 144;
    const int b  = tileM >> 10;                     // / (L/16)
    const int l0 = (tileM & 1023) << 4;
    const float* Xb = X + (size_t)b * CC * LL;

    v8f acc0 = vzero8();
    v8f acc1 = vzero8();
    #pragma unroll 4
    for (int k0 = 0; k0 < CC; k0 += 4) {
        const int ka = k0 + 2 * half;
        v2f a, w0, w1;
        a.x = Xb[(size_t)ka * LL + l0 + ml];
        a.y = Xb[(size_t)(ka + 1) * LL + l0 + ml];
        // prefetch next K-step of the streamed A operand (global_prefetch_b8)
        if (k0 + 4 < CC)
            __builtin_prefetch(&Xb[(size_t)(ka + 4) * LL + l0 + ml], 0, 1);
        w0.x = Wcat[ka * NPROJ + n0 + ml];
        w0.y = Wcat[(ka + 1) * NPROJ + n0 + ml];
        w1.x = Wcat[ka * NPROJ + n1 + ml];
        w1.y = Wcat[(ka + 1) * NPROJ + n1 + ml];
        acc0 = wmma4(a, w0, acc0);
        acc1 = wmma4(a, w1, acc1);
    }
    const float bias0 = bcat[n0 + ml];
    const float bias1 = bcat[n1 + ml];
    float* Yp0 = Y + ((size_t)(b * LL + l0)) * NPROJ + n0 + ml;
    float* Yp1 = Y + ((size_t)(b * LL + l0)) * NPROJ + n1 + ml;
    #pragma unroll
    for (int r = 0; r < 8; ++r) {
        const int m = r + 8 * half;
        Yp0[(size_t)m * NPROJ] = acc0[r] + bias0;
        Yp1[(size_t)m * NPROJ] = acc1[r] + bias1;
    }
}

// ------------------------------------------------------------------
// K3: deformable trilinear sampling. One wave per (b, l, head); lane = hd.
// Y row layout: [0:192] value, [192:264] offsets, [264:288] attn logits.
// ------------------------------------------------------------------
__global__ __launch_bounds__(256) void k_deform(
        const float* __restrict__ Y, float* __restrict__ ATT) {
    const int wid  = blockIdx.x * 8 + (threadIdx.x >> 5);
    const int lane = threadIdx.x & 31;
    const int b    = wid / (LL * NHD);
    const int rem  = wid % (LL * NHD);
    const int l    = rem / NHD;
    const int nh   = rem % NHD;

    const float* Yrow = Y + ((size_t)(b * LL + l)) * NPROJ;
    // softmax over P=4 attention logits
    float lg0 = Yrow[264 + nh * 4 + 0];
    float lg1 = Yrow[264 + nh * 4 + 1];
    float lg2 = Yrow[264 + nh * 4 + 2];
    float lg3 = Yrow[264 + nh * 4 + 3];
    float mx = fmaxf(fmaxf(lg0, lg1), fmaxf(lg2, lg3));
    float e0 = expf(lg0 - mx), e1 = expf(lg1 - mx),
          e2 = expf(lg2 - mx), e3 = expf(lg3 - mx);
    float inv = 1.f / (e0 + e1 + e2 + e3);
    float awv[4] = {e0 * inv, e1 * inv, e2 * inv, e3 * inv};

    // reference point for token l = d*H*W + y*W + x, normalized (d, x, y)
    const int d  = l >> 10;          // /(H*W)
    const int r2 = l & 1023;
    const int yy = r2 >> 5;
    const int xx = r2 & 31;
    const float refd = (d  + 0.5f) * (1.f / SS);
    const float refx = (xx + 0.5f) * (1.f / WW);
    const float refy = (yy + 0.5f) * (1.f / HH);

    const float* Yb = Y + (size_t)b * LL * NPROJ;
    float acc = 0.f;
    #pragma unroll
    for (int p = 0; p < PP; ++p) {
        const float* op = Yrow + 192 + (nh * PP + p) * 3;
        // pixel coords (align_corners=False)
        float pd = (refd + op[0] * (1.f / SS)) * SS - 0.5f;
        float px = (refx + op[1] * (1.f / WW)) * WW - 0.5f;
        float py = (refy + op[2] * (1.f / HH)) * HH - 0.5f;
        float d0f = floorf(pd), x0f = floorf(px), y0f = floorf(py);
        float fd = pd - d0f, fx = px - x0f, fy = py - y0f;
        int d0 = (int)d0f, x0 = (int)x0f, y0 = (int)y0f;
        const float wp = awv[p];
        #pragma unroll
        for (int dd = 0; dd < 2; ++dd)
        #pragma unroll
        for (int dy = 0; dy < 2; ++dy)
        #pragma unroll
        for (int dx = 0; dx < 2; ++dx) {
            int di = d0 + dd, yi = y0 + dy, xi = x0 + dx;
            float wgt = (dd ? fd : 1.f - fd) * (dy ? fy : 1.f - fy) * (dx ? fx : 1.f - fx);
            bool inb = (di >= 0) & (di < SS) & (yi >= 0) & (yi < HH) & (xi >= 0) & (xi < WW);
            int dc = min(max(di, 0), SS - 1);
            int yc = min(max(yi, 0), HH - 1);
            int xc = min(max(xi, 0), WW - 1);
            int idx = (dc * HH + yc) * WW + xc;
            float samp = Yb[(size_t)idx * NPROJ + nh * HD + lane];
            acc += (inb ? wgt : 0.f) * wp * samp;
        }
    }
    ATT[((size_t)(b * LL + l)) * CC + nh * HD + lane] = acc;
}

// ------------------------------------------------------------------
// K4: H = shortcut + ATT @ Wout + bout
// shortcut[b,l,c] = X[b*C*L + l*C + c]  (faithful channel-first view)
// ------------------------------------------------------------------
__global__ __launch_bounds__(256) void k_attn_out(
        const float* __restrict__ ATT, const float* __restrict__ Wout,
        const float* __restrict__ bout, const float* __restrict__ X,
        float* __restrict__ Hb) {
    __shared__ float aT[16 * 196];                  // padded stride 196 (bank-safe)
    const int tileM = blockIdx.x;                   // 0..2047
    const int b  = tileM >> 10;
    const int l0 = (tileM & 1023) << 4;
    const float* Ab = ATT + ((size_t)(b * LL + l0)) * CC;
    for (int i = threadIdx.x; i < 16 * CC; i += 256) {
        int m = i / CC, c = i % CC;
        aT[m * 196 + c] = Ab[m * CC + c];
    }
    __syncthreads();

    const int lane = threadIdx.x & 31;
    const int wv   = threadIdx.x >> 5;
    const int half = lane >> 4;
    const int ml   = lane & 15;
    const float* Sc = X + (size_t)b * CC * LL + (size_t)l0 * CC;  // shortcut tile
    for (int nt = wv; nt < 12; nt += 8) {
        const int n0 = nt * 16;
        v8f acc = vzero8();
        #pragma unroll 4
        for (int k0 = 0; k0 < CC; k0 += 4) {
            const int ka = k0 + 2 * half;
            v2f a, w;
            a.x = aT[ml * 196 + ka];
            a.y = aT[ml * 196 + ka + 1];
            w.x = Wout[ka * CC + n0 + ml];
            w.y = Wout[(ka + 1) * CC + n0 + ml];
            acc = wmma4(a, w, acc);
        }
        const float bias = bout[n0 + ml];
        float* Hp = Hb + ((size_t)(b * LL + l0)) * CC + n0 + ml;
        const float* Sp = Sc + n0 + ml;
        #pragma unroll
        for (int r = 0; r < 8; ++r) {
            const int m = r + 8 * half;
            Hp[m * CC] = acc[r] + bias + Sp[m * CC];
        }
    }
}

// ------------------------------------------------------------------
// K5: LN = layernorm(H) * g2 + b2   (one wave32 per token)
// ------------------------------------------------------------------
__global__ __launch_bounds__(256) void k_layernorm(
        const float* __restrict__ Hb, const float* __restrict__ g,
        const float* __restrict__ be, float* __restrict__ LN) {
    const int token = blockIdx.x * 8 + (threadIdx.x >> 5);
    const int lane  = threadIdx.x & 31;
    const float* row = Hb + (size_t)token * CC;
    float v[6], s = 0.f, s2 = 0.f;
    #pragma unroll
    for (int j = 0; j < 6; ++j) {
        v[j] = row[lane + 32 * j];
        s += v[j]; s2 += v[j] * v[j];
    }
    #pragma unroll
    for (int o = 16; o > 0; o >>= 1) {
        s  += __shfl_xor(s,  o, 32);
        s2 += __shfl_xor(s2, o, 32);
    }
    const float mean = s * (1.f / CC);
    const float var  = s2 * (1.f / CC) - mean * mean;
    const float rstd = rsqrtf(var + 1e-5f);
    float* orow = LN + (size_t)token * CC;
    #pragma unroll
    for (int j = 0; j < 6; ++j) {
        const int c = lane + 32 * j;
        orow[c] = (v[j] - mean) * rstd * g[c] + be[c];
    }
}

// ------------------------------------------------------------------
// K6: fused MLP + residual + transpose store into (B,C,S,H,W)
//   out[b,c,l] = H[b,l,c] + (gelu(LN @ W1 + b1) @ W2 + b2m)[l,c]
// 32-token tile per workgroup (halves W1/W2 L2 re-streaming vs 16);
// hidden (32x768) lives only in LDS, never touches HBM.
// ------------------------------------------------------------------
#define MT   32    // tokens per block
#define HSTR 772   // 768+4: 772 % 64 == 4 -> column reads hit 16 distinct banks
#define LNSTR 196  // 192+4
#define OSTR 33    // MT+1, gcd(33,64)=1 -> bank-safe column writes
__global__ __launch_bounds__(512) void k_mlp(
        const float* __restrict__ LN, const float* __restrict__ W1,
        const float* __restrict__ b1, const float* __restrict__ W2,
        const float* __restrict__ b2m, const float* __restrict__ Hb,
        float* __restrict__ Out) {
    // region0: lnT (MT x LNSTR = 6272) reused as oT (192 x OSTR = 6336)
    __shared__ float smem[CC * OSTR + MT * HSTR];   // 31040 floats = 121.3 KB
    float* lnT = smem;
    float* oT  = smem;
    float* hid = smem + CC * OSTR;

    const int tileM = blockIdx.x;                   // 0 .. B*L/MT - 1
    const int b  = tileM >> 9;                      // / (L/MT)
    const int l0 = (tileM & 511) * MT;

    const float* Lb = LN + ((size_t)(b * LL + l0)) * CC;
    for (int i = threadIdx.x; i < MT * CC; i += 512) {
        int m = i / CC, c = i % CC;
        lnT[m * LNSTR + c] = Lb[m * CC + c];
    }
    __syncthreads();

    const int lane = threadIdx.x & 31;
    const int wv   = threadIdx.x >> 5;
    const int half = lane >> 4;
    const int ml   = lane & 15;

    // stage 1: (32x192) @ (192x768) -> gelu -> LDS
    // 2 m-subtiles x 48 n-tiles = 96 wave-tiles over 16 waves
    for (int t = wv; t < 96; t += 16) {
        const int ms = (t & 1) * 16;
        const int n0 = (t >> 1) * 16;
        v8f acc = vzero8();
        #pragma unroll 4
        for (int k0 = 0; k0 < CC; k0 += 4) {
            const int ka = k0 + 2 * half;
            v2f a, w;
            a.x = lnT[(ms + ml) * LNSTR + ka];
            a.y = lnT[(ms + ml) * LNSTR + ka + 1];
            w.x = W1[ka * HID + n0 + ml];
            w.y = W1[(ka + 1) * HID + n0 + ml];
            acc = wmma4(a, w, acc);
        }
        const float bias = b1[n0 + ml];
        #pragma unroll
        for (int r = 0; r < 8; ++r) {
            const int m = ms + r + 8 * half;
            float x = acc[r] + bias;
            hid[m * HSTR + n0 + ml] =
                0.5f * x * (1.0f + erff(x * 0.70710678118654752f));  // exact gelu
        }
    }
    __syncthreads();

    // stage 2: (32x768) @ (768x192); 2 m-subtiles x 12 n-tiles = 24 wave-tiles
    for (int t = wv; t < 24; t += 16) {
        const int ms = (t & 1) * 16;
        const int n0 = (t >> 1) * 16;
        v8f acc = vzero8();
        #pragma unroll 4
        for (int k0 = 0; k0 < HID; k0 += 4) {
            const int ka = k0 + 2 * half;
            v2f a, w;
            a.x = hid[(ms + ml) * HSTR + ka];
            a.y = hid[(ms + ml) * HSTR + ka + 1];
            // prefetch the streamed W2 rows a few K-steps ahead
            if (k0 + 8 < HID)
                __builtin_prefetch(&W2[(ka + 8) * CC + n0 + ml], 0, 1);
            w.x = W2[ka * CC + n0 + ml];
            w.y = W2[(ka + 1) * CC + n0 + ml];
            acc = wmma4(a, w, acc);
        }
        #pragma unroll
        for (int r = 0; r < 8; ++r) {
            const int m = ms + r + 8 * half;
            oT[(n0 + ml) * OSTR + m] = acc[r];
        }
    }
    __syncthreads();

    // epilogue: residual + bias, coalesced transposed store (MT tokens per chan)
    const float* Hrow = Hb + ((size_t)(b * LL + l0)) * CC;
    float* Ob = Out + (size_t)b * CC * LL;
    for (int i = threadIdx.x; i < CC * MT; i += 512) {
        const int c = i / MT, m = i % MT;
        Ob[(size_t)c * LL + l0 + m] = oT[c * OSTR + m] + b2m[c] + Hrow[m * CC + c];
    }
}

// ------------------------------------------------------------------
extern "C" void kernel_launch(void* const* d_in, const int* in_sizes, int n_in,
                              void* d_out, int out_size, void* d_ws, size_t ws_size,
                              hipStream_t stream) {
    const float* X    = (const float*)d_in[0];
    // d_in[1] = mask_matrix (unused, shift_size == 0)
    const float* Wv   = (const float*)d_in[2];
    const float* bv   = (const float*)d_in[3];
    const float* Woff = (const float*)d_in[4];
    const float* boff = (const float*)d_in[5];
    const float* Waw  = (const float*)d_in[6];
    const float* baw  = (const float*)d_in[7];
    const float* Wout = (const float*)d_in[8];
    const float* bout = (const float*)d_in[9];
    const float* g2   = (const float*)d_in[10];
    const float* b2   = (const float*)d_in[11];
    const float* W1   = (const float*)d_in[12];
    const float* b1   = (const float*)d_in[13];
    const float* W2   = (const float*)d_in[14];
    const float* b2m  = (const float*)d_in[15];
    float* Out = (float*)d_out;

    // workspace layout (floats)
    float* ws   = (float*)d_ws;
    float* Wcat = ws;                          // 192*288          = 55296
    float* bcat = Wcat + CC * NPROJ;           // 288
    float* Y    = bcat + NPROJ;                // B*L*288          = 9437184
    float* ATT  = Y + (size_t)BB * LL * NPROJ; // B*L*192          = 6291456
    float* Hb   = ATT + (size_t)BB * LL * CC;  // B*L*192
    float* LN   = Hb + (size_t)BB * LL * CC;   // B*L*192

    (void)in_sizes; (void)n_in; (void)out_size; (void)ws_size;

    k_pack<<<(CC * NPROJ + 255) / 256, 256, 0, stream>>>(
        Wv, bv, Woff, boff, Waw, baw, Wcat, bcat);

    dim3 gProj(BB * LL / (16 * 8), NPROJ / 32);          // (256, 9): 2 N-tiles/wave
    k_gemm_proj<<<gProj, 256, 0, stream>>>(X, Wcat, bcat, Y);

    k_deform<<<BB * LL * NHD / 8, 256, 0, stream>>>(Y, ATT);

    k_attn_out<<<BB * LL / 16, 256, 0, stream>>>(ATT, Wout, bout, X, Hb);

    k_layernorm<<<BB * LL / 8, 256, 0, stream>>>(Hb, g2, b2, LN);

    k_mlp<<<BB * LL / MT, 512, 0, stream>>>(LN, W1, b1, W2, b2m, Hb, Out);
}